// ComplexDenseSO2_21869973471446
// MI455X (gfx1250) — compile-verified
//
#include <hip/hip_runtime.h>

// ComplexDenseSO2 on MI455X (gfx1250):
//   pass1: coeffs = (X . conj(bases)^T) * w   [B,K], reduce over D
//   pass2: out    = coeffs . bases            [B,D], reduce over K
// fp32 end-to-end via V_WMMA_F32_16X16X4_F32 (memory-bound: 2 streaming
// passes over 531 MB of bases ~= 1.06 GB -> ~46 us floor at 23.3 TB/s).
// Cache policy: bases is stream-once per pass (non-temporal loads, keep it
// out of the 192 MB L2); X (16.6 MB, ~64x reuse) and coeffs (1 MB, ~2025x
// reuse) stay default-temporal so they remain L2-resident; out is
// write-once (non-temporal stores).

typedef __attribute__((ext_vector_type(2))) float v2f;
typedef __attribute__((ext_vector_type(8))) float v8f;

#define WMMA_F32(a, b, c) \
    __builtin_amdgcn_wmma_f32_16x16x4_f32(false, (a), false, (b), (short)0, (c), false, false)

// Non-temporal streaming load of a float2 fragment (bases only).
static __device__ __forceinline__ v2f nt_load2(const float* p) {
    return __builtin_nontemporal_load((const v2f*)p);
}
static __device__ __forceinline__ float nt_load1(const float* p) {
    return __builtin_nontemporal_load(p);
}

namespace {
constexpr int Bb = 64;      // batch
constexpr int Kk = 2048;    // bases
constexpr int Dd = 32400;   // features
constexpr int CH = 12;      // D-chunks in pass 1
constexpr int STEPS1 = (Dd / 4) / CH;  // 675 k-steps of 4 per chunk
constexpr int BK = Bb * Kk;
}

// ---------------------------------------------------------------------------
// Pass 1: partial (X . conj(bases)^T) over one D-chunk.
// Block = 128 threads (4 waves). wave -> M-rows [16w,16w+16); block covers
// 32 basis columns (two 16-wide WMMA col tiles); grid = (K/32, CH).
// ---------------------------------------------------------------------------
__global__ __launch_bounds__(128) void
cds_pass1(const float* __restrict__ Xre, const float* __restrict__ Xim,
          const float* __restrict__ Bre, const float* __restrict__ Bim,
          float* __restrict__ partials)
{
    const int lane = threadIdx.x & 31;
    const int wave = threadIdx.x >> 5;
    const int h    = lane >> 4;     // half-wave: selects K pair {0,1} vs {2,3}
    const int l16  = lane & 15;

    const int m0    = wave * 16;            // batch-row tile
    const int n0    = blockIdx.x * 32;      // basis-column tile (2 subtiles)
    const int chunk = blockIdx.y;
    const int d0    = chunk * STEPS1 * 4;

    // A frag (16x4 of X): lane holds X[m0+l16][d + 2h + {0,1}] -> float2
    const float* ax_re = Xre + (size_t)(m0 + l16) * Dd + d0 + 2 * h;
    const float* ax_im = Xim + (size_t)(m0 + l16) * Dd + d0 + 2 * h;
    // B frag (4x16 of conj(bases)^T): lane holds bases[n0+l16][d + 2h + {0,1}]
    const float* b0_re = Bre + (size_t)(n0 + l16) * Dd + d0 + 2 * h;
    const float* b0_im = Bim + (size_t)(n0 + l16) * Dd + d0 + 2 * h;
    const float* b1_re = b0_re + (size_t)16 * Dd;
    const float* b1_im = b0_im + (size_t)16 * Dd;

    v8f accRe0 = {0,0,0,0,0,0,0,0}, accIm0 = {0,0,0,0,0,0,0,0};
    v8f accRe1 = {0,0,0,0,0,0,0,0}, accIm1 = {0,0,0,0,0,0,0,0};

    #pragma unroll 3
    for (int s = 0; s < STEPS1; ++s) {
        v2f axr = *(const v2f*)ax_re;
        v2f axi = *(const v2f*)ax_im;
        v2f b0r = nt_load2(b0_re);
        v2f b0i = nt_load2(b0_im);
        v2f b1r = nt_load2(b1_re);
        v2f b1i = nt_load2(b1_im);
        v2f naxr; naxr.x = -axr.x; naxr.y = -axr.y;

        // X . conj(b):  re = Xre*bre + Xim*bim ; im = Xim*bre - Xre*bim
        accRe0 = WMMA_F32(axr,  b0r, accRe0);
        accRe0 = WMMA_F32(axi,  b0i, accRe0);
        accIm0 = WMMA_F32(axi,  b0r, accIm0);
        accIm0 = WMMA_F32(naxr, b0i, accIm0);

        accRe1 = WMMA_F32(axr,  b1r, accRe1);
        accRe1 = WMMA_F32(axi,  b1i, accRe1);
        accIm1 = WMMA_F32(axi,  b1r, accIm1);
        accIm1 = WMMA_F32(naxr, b1i, accIm1);

        ax_re += 4; ax_im += 4;
        b0_re += 4; b0_im += 4;
        b1_re += 4; b1_im += 4;
    }

    // C layout: VGPR r -> M = r (lanes 0-15) / 8+r (lanes 16-31), N = l16.
    float* pre = partials + (size_t)chunk * 2 * BK;
    float* pim = pre + BK;
    #pragma unroll
    for (int r = 0; r < 8; ++r) {
        const int row = m0 + r + 8 * h;
        const int idx = row * Kk + n0 + l16;
        pre[idx]      = accRe0[r];
        pim[idx]      = accIm0[r];
        pre[idx + 16] = accRe1[r];
        pim[idx + 16] = accIm1[r];
    }
}

// ---------------------------------------------------------------------------
// Pass 1b: reduce chunk partials, apply complex weight w[k] -> coeff planes.
// ---------------------------------------------------------------------------
__global__ __launch_bounds__(256) void
cds_scale(const float* __restrict__ partials,
          const float* __restrict__ wre, const float* __restrict__ wim,
          float* __restrict__ cre, float* __restrict__ cim)
{
    const int tid = blockIdx.x * 256 + threadIdx.x;   // < B*K
    const int k   = tid & (Kk - 1);
    float sre = 0.f, sim = 0.f;
    #pragma unroll
    for (int c = 0; c < CH; ++c) {
        const float* p = partials + (size_t)c * 2 * BK;
        sre += p[tid];
        sim += p[tid + BK];
    }
    const float wr = wre[k], wi = wim[k];
    cre[tid] = sre * wr - sim * wi;
    cim[tid] = sre * wi + sim * wr;
}

// ---------------------------------------------------------------------------
// Pass 2: out[B,D] = coeffs . bases (complex), interleaved complex64 output.
// Block = 128 threads (4 waves = 4 M tiles), block covers 16 d-columns.
// grid = D/16 = 2025 (exact, no tail).
// ---------------------------------------------------------------------------
__global__ __launch_bounds__(128) void
cds_pass2(const float* __restrict__ cre, const float* __restrict__ cim,
          const float* __restrict__ Bre, const float* __restrict__ Bim,
          float* __restrict__ out)
{
    const int lane = threadIdx.x & 31;
    const int wave = threadIdx.x >> 5;
    const int h    = lane >> 4;
    const int l16  = lane & 15;
    const int m0   = wave * 16;
    const int d0   = blockIdx.x * 16;

    // A frag: coeffs[m0+l16][kk + 2h + {0,1}] -> float2 (K contiguous)
    const float* a_re = cre + (size_t)(m0 + l16) * Kk + 2 * h;
    const float* a_im = cim + (size_t)(m0 + l16) * Kk + 2 * h;
    // B frag: bases[kk + 2h + v][d0 + l16]; two coalesced dword loads per frag
    const float* b_re0 = Bre + (size_t)(2 * h)     * Dd + d0 + l16;
    const float* b_re1 = Bre + (size_t)(2 * h + 1) * Dd + d0 + l16;
    const float* b_im0 = Bim + (size_t)(2 * h)     * Dd + d0 + l16;
    const float* b_im1 = Bim + (size_t)(2 * h + 1) * Dd + d0 + l16;

    v8f accRe = {0,0,0,0,0,0,0,0}, accIm = {0,0,0,0,0,0,0,0};

    #pragma unroll 4
    for (int s = 0; s < Kk / 4; ++s) {
        v2f ar = *(const v2f*)a_re;
        v2f ai = *(const v2f*)a_im;
        v2f nai; nai.x = -ai.x; nai.y = -ai.y;
        v2f br, bi;
        br.x = nt_load1(b_re0); br.y = nt_load1(b_re1);
        bi.x = nt_load1(b_im0); bi.y = nt_load1(b_im1);

        // c*b: re = cre*bre - cim*bim ; im = cre*bim + cim*bre
        accRe = WMMA_F32(ar,  br, accRe);
        accRe = WMMA_F32(nai, bi, accRe);
        accIm = WMMA_F32(ar,  bi, accIm);
        accIm = WMMA_F32(ai,  br, accIm);

        a_re += 4; a_im += 4;
        b_re0 += (size_t)4 * Dd; b_re1 += (size_t)4 * Dd;
        b_im0 += (size_t)4 * Dd; b_im1 += (size_t)4 * Dd;
    }

    // Interleaved complex store: out[(row*D + col)*2] = {re, im}; write-once.
    #pragma unroll
    for (int r = 0; r < 8; ++r) {
        const int row = m0 + r + 8 * h;
        const size_t idx = ((size_t)row * Dd + d0 + l16) * 2;
        v2f o; o.x = accRe[r]; o.y = accIm[r];
        __builtin_nontemporal_store(o, (v2f*)(out + idx));
    }
}

// ---------------------------------------------------------------------------
extern "C" void kernel_launch(void* const* d_in, const int* in_sizes, int n_in,
                              void* d_out, int out_size, void* d_ws, size_t ws_size,
                              hipStream_t stream)
{
    const float* Xre = (const float*)d_in[0];
    const float* Xim = (const float*)d_in[1];
    const float* Bre = (const float*)d_in[2];
    const float* Bim = (const float*)d_in[3];
    const float* wre = (const float*)d_in[4];
    const float* wim = (const float*)d_in[5];
    float* out = (float*)d_out;

    // Workspace layout: [cre BK][cim BK][partials CH*2*BK]  (~13.6 MB)
    float* cre      = (float*)d_ws;
    float* cim      = cre + BK;
    float* partials = cim + BK;

    dim3 g1(Kk / 32, CH);
    cds_pass1<<<g1, 128, 0, stream>>>(Xre, Xim, Bre, Bim, partials);
    cds_scale<<<BK / 256, 256, 0, stream>>>(partials, wre, wim, cre, cim);
    cds_pass2<<<Dd / 16, 128, 0, stream>>>(cre, cim, Bre, Bim, out);
}